// Cut_Balance_Loss_28578712388223
// MI455X (gfx1250) — compile-verified
//
#include <hip/hip_runtime.h>

// ---------------------------------------------------------------------------
// Cut/Balance loss for MI455X (gfx1250, wave32).
//
// ws layout (floats):
//   [0..15]                      : column sums of Y (atomicAdd targets)
//   [16 .. 16+EDGE_BLOCKS-1]     : per-block partials of sum(Y[src]*(1-Y[dst]))
//   [16+EB .. 16+2*EB-1]         : per-block partials of sum(A_vals)
// ---------------------------------------------------------------------------

typedef __attribute__((ext_vector_type(2))) float v2f;
typedef __attribute__((ext_vector_type(8))) float v8f;

#define TPB 256
#define EDGE_BLOCKS 1024
#define COL_BLOCKS 128

__global__ void init_ws(float* __restrict__ ws, int n) {
    int i = blockIdx.x * blockDim.x + threadIdx.x;
    for (; i < n; i += gridDim.x * blockDim.x) ws[i] = 0.0f;
}

// Fused: Gamma partial (sum of A_vals) + edge-dot partial. Memory bound;
// Y rows are 64B and L2-resident, so per-lane float4 gathers are the right path.
__global__ void edge_reduce(const float4* __restrict__ Yv,
                            const float* __restrict__ A,
                            const int* __restrict__ src,
                            const int* __restrict__ dst,
                            float* __restrict__ dotPart,
                            float* __restrict__ aPart,
                            int E) {
    __shared__ float sdot[TPB];
    __shared__ float sa[TPB];
    const int tid = threadIdx.x;
    int e = blockIdx.x * blockDim.x + tid;
    const int stride = gridDim.x * blockDim.x;

    float dacc = 0.0f, aacc = 0.0f;
    for (; e < E; e += stride) {
        aacc += A[e];
        const int s = src[e];
        const int d = dst[e];
        const float4* __restrict__ ys = Yv + (size_t)s * 4;
        const float4* __restrict__ yd = Yv + (size_t)d * 4;
#pragma unroll
        for (int k = 0; k < 4; ++k) {
            const float4 a4 = ys[k];
            const float4 b4 = yd[k];
            dacc = fmaf(a4.x, 1.0f - b4.x, dacc);
            dacc = fmaf(a4.y, 1.0f - b4.y, dacc);
            dacc = fmaf(a4.z, 1.0f - b4.z, dacc);
            dacc = fmaf(a4.w, 1.0f - b4.w, dacc);
        }
    }

    sdot[tid] = dacc;
    sa[tid]   = aacc;
    __syncthreads();
    for (int s2 = TPB / 2; s2 > 0; s2 >>= 1) {
        if (tid < s2) {
            sdot[tid] += sdot[tid + s2];
            sa[tid]   += sa[tid + s2];
        }
        __syncthreads();
    }
    if (tid == 0) {
        dotPart[blockIdx.x] = sdot[0];
        aPart[blockIdx.x]   = sa[0];
    }
}

// Column sums of Y [N x 16] via V_WMMA_F32_16X16X4_F32 with an all-ones
// A-matrix: D = ones(16x4) * Y[base:base+4, 0:16] + C accumulates the column
// sums into every row of the 16x16 f32 accumulator.
//
// B (4x16, 32-bit) layout: row (K) striped across the two src VGPRs within a
// lane; lanes 0-15 carry K=0 (vgpr0) / K=1 (vgpr1) at N=lane, lanes 16-31
// carry K=2 / K=3 at N=lane-16.
// C/D layout: VGPR0 = {lanes 0-15: M=0,N=lane ; lanes 16-31: M=8,N=lane-16},
// so after the loop c[0] on lanes 0-15 is colsum[lane].
__global__ void colsum_wmma(const float* __restrict__ Y,
                            float* __restrict__ colsums, int N) {
    const int lane          = threadIdx.x & 31;
    const int waveInBlock   = threadIdx.x >> 5;
    const int wavesPerBlock = blockDim.x >> 5;
    const int gwave  = blockIdx.x * wavesPerBlock + waveInBlock;
    const int nwaves = gridDim.x * wavesPerBlock;

    v2f a;  a.x = 1.0f; a.y = 1.0f;   // all-ones A: layout-invariant
    v8f c = {};

    const int col     = lane & 15;
    const int rowoff0 = (lane < 16) ? 0 : 2;  // K held in B vgpr0 for this lane

    // Uniform trip count per wave; EXEC stays all-ones around the WMMA.
    for (int base = gwave * 4; base < N; base += nwaves * 4) {
        int r0 = base + rowoff0;
        int r1 = r0 + 1;
        int rc0 = (r0 < N) ? r0 : 0;          // v_cndmask guards, no divergence
        int rc1 = (r1 < N) ? r1 : 0;
        float b0 = Y[rc0 * 16 + col];
        float b1 = Y[rc1 * 16 + col];
        if (r0 >= N) b0 = 0.0f;
        if (r1 >= N) b1 = 0.0f;
        v2f b;  b.x = b0; b.y = b1;
        c = __builtin_amdgcn_wmma_f32_16x16x4_f32(
            /*neg_a=*/false, a, /*neg_b=*/false, b,
            /*c_mod=*/(short)0, c, /*reuse_a=*/false, /*reuse_b=*/false);
    }

    if (lane < 16) atomicAdd(&colsums[lane], c[0]);
}

__global__ void final_combine(const float* __restrict__ dotPart,
                              const float* __restrict__ aPart,
                              const float* __restrict__ colsums,
                              float* __restrict__ out,
                              int nblk, float target) {
    __shared__ float sdot[TPB];
    __shared__ float sa[TPB];
    const int tid = threadIdx.x;
    float d = 0.0f, a = 0.0f;
    for (int i = tid; i < nblk; i += TPB) {
        d += dotPart[i];
        a += aPart[i];
    }
    sdot[tid] = d;
    sa[tid]   = a;
    __syncthreads();
    for (int s2 = TPB / 2; s2 > 0; s2 >>= 1) {
        if (tid < s2) {
            sdot[tid] += sdot[tid + s2];
            sa[tid]   += sa[tid + s2];
        }
        __syncthreads();
    }
    if (tid == 0) {
        const float Gamma = sa[0];
        const float loss1 = sdot[0] / Gamma;
        float l2 = 0.0f;
#pragma unroll
        for (int j = 0; j < 16; ++j) {
            const float t = colsums[j] - target;
            l2 = fmaf(t, t, l2);
        }
        out[0] = loss1 + l2;   // loss
        out[1] = loss1;        // loss_1
        out[2] = l2;           // loss_2
    }
}

extern "C" void kernel_launch(void* const* d_in, const int* in_sizes, int n_in,
                              void* d_out, int out_size, void* d_ws, size_t ws_size,
                              hipStream_t stream) {
    const float* Y   = (const float*)d_in[0];
    const float* A   = (const float*)d_in[1];
    const int*   src = (const int*)d_in[2];
    const int*   dst = (const int*)d_in[3];

    const int NG = in_sizes[0];     // N * 16
    const int N  = NG / 16;
    const int E  = in_sizes[1];

    float* out = (float*)d_out;
    float* ws  = (float*)d_ws;
    float* colsums = ws;
    float* dotPart = ws + 16;
    float* aPart   = ws + 16 + EDGE_BLOCKS;

    int eblk = (E + TPB - 1) / TPB;
    if (eblk > EDGE_BLOCKS) eblk = EDGE_BLOCKS;

    init_ws<<<1, TPB, 0, stream>>>(ws, 16 + 2 * EDGE_BLOCKS);
    edge_reduce<<<eblk, TPB, 0, stream>>>((const float4*)Y, A, src, dst,
                                          dotPart, aPart, E);
    colsum_wmma<<<COL_BLOCKS, TPB, 0, stream>>>(Y, colsums, N);
    final_combine<<<1, TPB, 0, stream>>>(dotPart, aPart, colsums, out,
                                         eblk, (float)N / 16.0f);
    // Output Y (4th tuple element) directly after the 3 scalars.
    hipMemcpyAsync(out + 3, Y, (size_t)NG * sizeof(float),
                   hipMemcpyDeviceToDevice, stream);
}